// GAT_8057358648126
// MI455X (gfx1250) — compile-verified
//
#include <hip/hip_runtime.h>
#include <math.h>

// ---------------- problem constants (match reference) ----------------
#define NNODES 100000
#define NEDGES 1600000
#define EPLUS  (NEDGES + NNODES)   // edges + self loops = 1,700,000
#define FIN    128
#define NH     4                   // heads (all layers, incl. last)
#define CH     16                  // per-head channels, layers 1-2
#define DHID   64                  // NH*CH
#define NCLS   40                  // per-head channels layer 3 / output dim
#define NEG    0.2f
#define LNEPS  1e-5f

typedef float v2f __attribute__((ext_vector_type(2)));
typedef float v8f __attribute__((ext_vector_type(8)));

// ---------------- f32 WMMA GEMM: C[rows,COLS] = A[rows,K] @ B[K,COLS] (+bias) ---
// One wave32 per 16x16 output tile, V_WMMA_F32_16X16X4_F32, fully unrolled K.
// A layout per ISA: lane(0-15)=row M, VGPR i holds K = k0 + 2*half + i.
// B/C/D: row striped across lanes within a VGPR (halves offset by 2 K / 8 M).
// NOTE: rows must be a multiple of 16 (100000 = 6250*16). Out-of-range B columns
// are address-clamped; they only pollute their own (unstored) output column.
template<int K, int COLS>
__global__ void wmma_gemm_f32(const float* __restrict__ A, const float* __restrict__ B,
                              const float* __restrict__ bias1, const float* __restrict__ bias2,
                              float* __restrict__ C, int rows, int rtiles)
{
    constexpr int CT = (COLS + 15) / 16;
    int wid  = (blockIdx.x * blockDim.x + threadIdx.x) >> 5;   // global wave id
    if (wid >= rtiles * CT) return;                            // wave-uniform exit
    int lane = threadIdx.x & 31;
    int rowTile = wid % rtiles;
    int colTile = wid / rtiles;                                // consecutive waves share B tile
    int m0 = rowTile << 4, n0 = colTile << 4;
    int half = lane >> 4, lm = lane & 15;

    int arow = m0 + lm;                                        // always < rows (rows%16==0)
    const float* Ap = A + (size_t)arow * K + (half << 1);

    int bcolRaw = n0 + lm;
    bool bok = true;
    int bcol = bcolRaw;
    if constexpr (COLS % 16 != 0) {                            // only the COLS=40 case
        bok  = bcolRaw < COLS;
        bcol = bok ? bcolRaw : (COLS - 1);                     // clamp address, no branchy loads
    }
    const float* Bp = B + (size_t)(half << 1) * COLS + bcol;

    v8f acc0 = {0.f,0.f,0.f,0.f,0.f,0.f,0.f,0.f};
    v8f acc1 = {0.f,0.f,0.f,0.f,0.f,0.f,0.f,0.f};
#pragma unroll
    for (int k = 0; k < K; k += 8) {
        v2f a0 = *(const v2f*)(Ap + k);                        // global_load_b64
        v2f b0; b0[0] = Bp[(size_t)(k)     * COLS];
                b0[1] = Bp[(size_t)(k + 1) * COLS];
        acc0 = __builtin_amdgcn_wmma_f32_16x16x4_f32(false, a0, false, b0,
                                                     (short)0, acc0, false, false);
        v2f a1 = *(const v2f*)(Ap + k + 4);
        v2f b1; b1[0] = Bp[(size_t)(k + 4) * COLS];
                b1[1] = Bp[(size_t)(k + 5) * COLS];
        acc1 = __builtin_amdgcn_wmma_f32_16x16x4_f32(false, a1, false, b1,
                                                     (short)0, acc1, false, false);
    }
    v8f acc = acc0 + acc1;

    if (bok) {
        float bb = 0.f;
        if (bias1) bb += bias1[bcolRaw];
        if (bias2) bb += bias2[bcolRaw];
#pragma unroll
        for (int v = 0; v < 8; ++v) {
            int r = m0 + v + (half << 3);
            C[(size_t)r * COLS + bcolRaw] = acc[v] + bb;
        }
    }
    (void)rows;
}

// ---------------- attention helpers ----------------
__device__ __forceinline__ void atomicMaxF(float* addr, float val) {
    if (val >= 0.f) atomicMax((int*)addr, __float_as_int(val));
    else            atomicMin((unsigned int*)addr, __float_as_uint(val));
}

__device__ __forceinline__ void edge_sd(int e, const int* __restrict__ ei, int& s_, int& d_) {
    if (e < NEDGES) { s_ = ei[e]; d_ = ei[NEDGES + e]; }
    else            { s_ = d_ = e - NEDGES; }     // self loops appended
}

__device__ __forceinline__ float lrelu(float x) { return x > 0.f ? x : NEG * x; }

// a_s[n,h] = <h[n,h,:], a_src[h,:]>, a_d likewise. One thread per (node, head).
template<int C>
__global__ void attn_scores(const float* __restrict__ feat, const float* __restrict__ asrc,
                            const float* __restrict__ adst, float* __restrict__ as_,
                            float* __restrict__ ad_)
{
    int t = blockIdx.x * blockDim.x + threadIdx.x;
    if (t >= NNODES * NH) return;
    int n = t >> 2, h = t & 3;
    const float* f  = feat + (size_t)n * (NH * C) + h * C;
    const float* vs = asrc + h * C;
    const float* vd = adst + h * C;
    float ss = 0.f, sd = 0.f;
#pragma unroll
    for (int c = 0; c < C; ++c) { float x = f[c]; ss += x * vs[c]; sd += x * vd[c]; }
    as_[t] = ss; ad_[t] = sd;
}

__global__ void init_ms(float* __restrict__ m, float* __restrict__ s) {
    int t = blockIdx.x * blockDim.x + threadIdx.x;
    if (t < NNODES * NH) { m[t] = -INFINITY; s[t] = 0.f; }
}

// segment max of leaky-relu logits, grouped by dst
__global__ void edge_max(const int* __restrict__ ei, const float* __restrict__ as_,
                         const float* __restrict__ ad_, float* __restrict__ m)
{
    int e = blockIdx.x * blockDim.x + threadIdx.x;
    if (e >= EPLUS) return;
    int s_, d_; edge_sd(e, ei, s_, d_);
#pragma unroll
    for (int h = 0; h < NH; ++h)
        atomicMaxF(&m[d_ * NH + h], lrelu(as_[s_ * NH + h] + ad_[d_ * NH + h]));
}

// segment sum of exp(logit - max)
__global__ void edge_expsum(const int* __restrict__ ei, const float* __restrict__ as_,
                            const float* __restrict__ ad_, const float* __restrict__ m,
                            float* __restrict__ s)
{
    int t = blockIdx.x * blockDim.x + threadIdx.x;
    if (t >= EPLUS * NH) return;
    int e = t >> 2, h = t & 3;
    int s_, d_; edge_sd(e, ei, s_, d_);
    float el = lrelu(as_[s_ * NH + h] + ad_[d_ * NH + h]);
    atomicAdd(&s[d_ * NH + h], __expf(el - m[d_ * NH + h]));
}

// alpha = exp(logit - max) / sum ; out[dst] += scale * alpha * feat[src]
// CONCAT: out stride NH*C with head offset. else: out stride C (mean over heads via scale).
template<int C, bool CONCAT>
__global__ void edge_aggregate(const int* __restrict__ ei, const float* __restrict__ as_,
                               const float* __restrict__ ad_, const float* __restrict__ m,
                               const float* __restrict__ s, const float* __restrict__ feat,
                               float* __restrict__ out, float scale,
                               float* __restrict__ alphaOut)
{
    int t = blockIdx.x * blockDim.x + threadIdx.x;
    if (t >= EPLUS * NH) return;
    int e = t >> 2, h = t & 3;
    int s_, d_; edge_sd(e, ei, s_, d_);
    float el    = lrelu(as_[s_ * NH + h] + ad_[d_ * NH + h]);
    float alpha = __expf(el - m[d_ * NH + h]) / s[d_ * NH + h];
    if (alphaOut) alphaOut[(size_t)e * NH + h] = alpha;
    const float* f = feat + (size_t)s_ * (NH * C) + h * C;
    float* o = CONCAT ? (out + (size_t)d_ * (NH * C) + h * C)
                      : (out + (size_t)d_ * C);
    float w = alpha * scale;
#pragma unroll
    for (int c = 0; c < C; ++c) atomicAdd(&o[c], w * f[c]);
}

// ---------------- LayerNorm (+ optional ELU), one wave32 per node -------------
template<int DIM, bool ELU>
__global__ void ln_kernel(float* __restrict__ buf, const float* __restrict__ g,
                          const float* __restrict__ b)
{
    int wid  = (blockIdx.x * blockDim.x + threadIdx.x) >> 5;
    if (wid >= NNODES) return;
    int lane = threadIdx.x & 31;
    float* row = buf + (size_t)wid * DIM;
    float x0 = (lane < DIM)      ? row[lane]      : 0.f;
    float x1 = (lane + 32 < DIM) ? row[lane + 32] : 0.f;
    float sum = x0 + x1;
    float sq  = x0 * x0 + x1 * x1;
#pragma unroll
    for (int off = 16; off >= 1; off >>= 1) {
        sum += __shfl_xor(sum, off, 32);
        sq  += __shfl_xor(sq,  off, 32);
    }
    float mu  = sum / DIM;
    float var = sq / DIM - mu * mu;
    float r   = rsqrtf(var + LNEPS);
    if (lane < DIM) {
        float y = (x0 - mu) * r * g[lane] + b[lane];
        if (ELU) y = y > 0.f ? y : __expf(y) - 1.f;
        row[lane] = y;
    }
    if (lane + 32 < DIM) {
        float y = (x1 - mu) * r * g[lane + 32] + b[lane + 32];
        if (ELU) y = y > 0.f ? y : __expf(y) - 1.f;
        row[lane + 32] = y;
    }
}

// ---------------- host-side orchestration ----------------
static inline int cdiv(long long a, long long b) { return (int)((a + b - 1) / b); }

extern "C" void kernel_launch(void* const* d_in, const int* in_sizes, int n_in,
                              void* d_out, int out_size, void* d_ws, size_t ws_size,
                              hipStream_t stream)
{
    (void)in_sizes; (void)n_in; (void)out_size; (void)ws_size;

    // inputs in setup_inputs() flattened dict order
    const float* x       = (const float*)d_in[0];   // [N,128]
    const int*   ei      = (const int*)  d_in[1];   // [2,E]
    const float* Wc1     = (const float*)d_in[2];   // conv_in.W   [128,64]
    const float* aS1     = (const float*)d_in[3];   // conv_in.a_src [4,16]
    const float* aD1     = (const float*)d_in[4];
    const float* bc1     = (const float*)d_in[5];   // conv_in.b [64]
    const float* Wl1     = (const float*)d_in[6];   // lin_in.W [128,64]
    const float* bl1     = (const float*)d_in[7];
    const float* g1      = (const float*)d_in[8];
    const float* be1     = (const float*)d_in[9];
    const float* Wc2     = (const float*)d_in[10];  // conv_mid.W [64,64]
    const float* aS2     = (const float*)d_in[11];
    const float* aD2     = (const float*)d_in[12];
    const float* bc2     = (const float*)d_in[13];
    const float* Wl2     = (const float*)d_in[14];  // lin_mid.W [64,64]
    const float* bl2     = (const float*)d_in[15];
    const float* g2      = (const float*)d_in[16];
    const float* be2     = (const float*)d_in[17];
    const float* Wc3     = (const float*)d_in[18];  // conv_out.W [64,160]
    const float* aS3     = (const float*)d_in[19];  // [4,40]
    const float* aD3     = (const float*)d_in[20];
    const float* bc3     = (const float*)d_in[21];  // [40]
    const float* Wl3     = (const float*)d_in[22];  // lin_out.W [64,40]
    const float* bl3     = (const float*)d_in[23];
    const float* g3      = (const float*)d_in[24];
    const float* be3     = (const float*)d_in[25];

    float* out   = (float*)d_out;                   // [N,40]
    float* alpha = out + (size_t)NNODES * NCLS;     // [EPLUS,4]

    // workspace layout (floats)
    float* ws   = (float*)d_ws;
    float* feat = ws;                               // [N,160] conv features (all layers)
    float* x1   = feat + (size_t)NNODES * 160;      // [N,64] layer1 out / layer2 in
    float* x2   = x1   + (size_t)NNODES * DHID;     // [N,64] layer2 out / layer3 in
    float* as_  = x2   + (size_t)NNODES * DHID;     // [N,4]
    float* ad_  = as_  + (size_t)NNODES * NH;
    float* mbuf = ad_  + (size_t)NNODES * NH;
    float* sbuf = mbuf + (size_t)NNODES * NH;

    const int TPB = 256;
    const int RT  = cdiv(NNODES, 16);                 // 6250 row tiles
    const int gNH   = cdiv((long long)NNODES * NH, TPB);
    const int gE    = cdiv(EPLUS, TPB);
    const int gEH   = cdiv((long long)EPLUS * NH, TPB);
    const int gNode = cdiv((long long)NNODES * 32, TPB);   // wave per node

    auto gblocks = [&](int ctiles) {
        return cdiv((long long)RT * ctiles * 32, TPB);
    };

    // ---------------- layer 1 ----------------
    wmma_gemm_f32<FIN, DHID><<<gblocks(4), TPB, 0, stream>>>(
        x, Wc1, nullptr, nullptr, feat, NNODES, RT);                 // conv features
    wmma_gemm_f32<FIN, DHID><<<gblocks(4), TPB, 0, stream>>>(
        x, Wl1, bl1, bc1, x1, NNODES, RT);                           // skip + biases
    attn_scores<CH><<<gNH, TPB, 0, stream>>>(feat, aS1, aD1, as_, ad_);
    init_ms<<<gNH, TPB, 0, stream>>>(mbuf, sbuf);
    edge_max<<<gE, TPB, 0, stream>>>(ei, as_, ad_, mbuf);
    edge_expsum<<<gEH, TPB, 0, stream>>>(ei, as_, ad_, mbuf, sbuf);
    edge_aggregate<CH, true><<<gEH, TPB, 0, stream>>>(ei, as_, ad_, mbuf, sbuf,
                                                      feat, x1, 1.0f, nullptr);
    ln_kernel<DHID, true><<<gNode, TPB, 0, stream>>>(x1, g1, be1);

    // ---------------- layer 2 ----------------
    wmma_gemm_f32<DHID, DHID><<<gblocks(4), TPB, 0, stream>>>(
        x1, Wc2, nullptr, nullptr, feat, NNODES, RT);
    wmma_gemm_f32<DHID, DHID><<<gblocks(4), TPB, 0, stream>>>(
        x1, Wl2, bl2, bc2, x2, NNODES, RT);
    attn_scores<CH><<<gNH, TPB, 0, stream>>>(feat, aS2, aD2, as_, ad_);
    init_ms<<<gNH, TPB, 0, stream>>>(mbuf, sbuf);
    edge_max<<<gE, TPB, 0, stream>>>(ei, as_, ad_, mbuf);
    edge_expsum<<<gEH, TPB, 0, stream>>>(ei, as_, ad_, mbuf, sbuf);
    edge_aggregate<CH, true><<<gEH, TPB, 0, stream>>>(ei, as_, ad_, mbuf, sbuf,
                                                      feat, x2, 1.0f, nullptr);
    ln_kernel<DHID, true><<<gNode, TPB, 0, stream>>>(x2, g2, be2);

    // ---------------- layer 3 (mean over heads, alpha exported) ----------------
    wmma_gemm_f32<DHID, NH * NCLS><<<gblocks(10), TPB, 0, stream>>>(
        x2, Wc3, nullptr, nullptr, feat, NNODES, RT);                // [N,160]
    wmma_gemm_f32<DHID, NCLS><<<gblocks(3), TPB, 0, stream>>>(
        x2, Wl3, bl3, bc3, out, NNODES, RT);                         // skip into d_out
    attn_scores<NCLS><<<gNH, TPB, 0, stream>>>(feat, aS3, aD3, as_, ad_);
    init_ms<<<gNH, TPB, 0, stream>>>(mbuf, sbuf);
    edge_max<<<gE, TPB, 0, stream>>>(ei, as_, ad_, mbuf);
    edge_expsum<<<gEH, TPB, 0, stream>>>(ei, as_, ad_, mbuf, sbuf);
    edge_aggregate<NCLS, false><<<gEH, TPB, 0, stream>>>(ei, as_, ad_, mbuf, sbuf,
                                                         feat, out, 1.0f / NH, alpha);
    ln_kernel<NCLS, false><<<gNode, TPB, 0, stream>>>(out, g3, be3);
}